// BoundaryLoss_34119220199972
// MI455X (gfx1250) — compile-verified
//
#include <hip/hip_runtime.h>
#include <math.h>

#define BATCH 64
#define H 512
#define W 512
#define TILE_H 8
#define LROWS (TILE_H + 2)                 // 10 rows incl. vertical halo
#define TILES_PER_IMG (H / TILE_H)         // 64
#define NBLOCKS (BATCH * TILES_PER_IMG)    // 4096
#define NTHREADS 256
#define LDS_ELEMS (LROWS * W)              // 5120 floats per buffer
#define CHUNKS (LDS_ELEMS / (NTHREADS * 4))// 5 float4 chunks per thread

// ---- CDNA5 async global->LDS copy (GVS mode: saddr base + 32-bit voffset) ----
__device__ __forceinline__ void async_load_b128(unsigned lds_byte_off,
                                                unsigned gbl_byte_off,
                                                const float* sbase) {
    asm volatile("global_load_async_to_lds_b128 %0, %1, %2"
                 :
                 : "v"(lds_byte_off), "v"(gbl_byte_off),
                   "s"((unsigned long long)(size_t)sbase)
                 : "memory");
}

__device__ __forceinline__ void wait_asynccnt0() {
    asm volatile("s_wait_asynccnt 0" ::: "memory");
}

// Low 32 bits of a generic pointer to LDS == wave-relative LDS byte offset
__device__ __forceinline__ unsigned lds_base_off(const void* p) {
    return (unsigned)(unsigned long long)(size_t)p;
}

__global__ __launch_bounds__(NTHREADS)
void sobel_bce_tile_kernel(const float* __restrict__ p,
                           const float* __restrict__ t,
                           float* __restrict__ partials) {
    __shared__ __align__(16) float sP[LDS_ELEMS];
    __shared__ __align__(16) float sT[LDS_ELEMS];
    __shared__ float swave[NTHREADS / 32];

    const int tid  = threadIdx.x;
    const int img  = blockIdx.x / TILES_PER_IMG;
    const int tile = blockIdx.x - img * TILES_PER_IMG;
    const int r0   = tile * TILE_H;

    const float* pimg = p + (size_t)img * (H * W);
    const float* timg = t + (size_t)img * (H * W);

    const unsigned ldsP0 = lds_base_off(sP);
    const unsigned ldsT0 = lds_base_off(sT);

    // ---------------- Phase A: async stage 10x512 tiles of p and t into LDS ----
    #pragma unroll
    for (int k = 0; k < CHUNKS; ++k) {
        const int e    = (k * NTHREADS + tid) * 4;   // flat float index in tile
        const int row  = e >> 9;                     // /512
        const int col  = e & (W - 1);
        const int grow = r0 - 1 + row;               // global image row
        if (grow >= 0 && grow < H) {
            const unsigned goff = (unsigned)((grow << 9) + col) * 4u;
            async_load_b128(ldsP0 + (unsigned)e * 4u, goff, pimg);
            async_load_b128(ldsT0 + (unsigned)e * 4u, goff, timg);
        } else {
            // zero padding ('SAME' conv) for halo rows outside the image
            float4 z = make_float4(0.f, 0.f, 0.f, 0.f);
            *(float4*)&sP[e] = z;
            *(float4*)&sT[e] = z;
        }
    }
    wait_asynccnt0();
    __syncthreads();

    // ---------------- Phase B: sigmoid in place, once per element -------------
    #pragma unroll
    for (int k = 0; k < CHUNKS; ++k) {
        const int e = (k * NTHREADS + tid) * 4;
        float4 v = *(const float4*)&sP[e];
        v.x = 1.f / (1.f + __expf(-v.x));
        v.y = 1.f / (1.f + __expf(-v.y));
        v.z = 1.f / (1.f + __expf(-v.z));
        v.w = 1.f / (1.f + __expf(-v.w));
        *(float4*)&sP[e] = v;
    }
    __syncthreads();

    // ---------------- Phase C: 3x3 Sobel stencil + loss, 4 strips of 1x4 ------
    float acc = 0.f;
    #pragma unroll
    for (int s = 0; s < (TILE_H * W / 4) / NTHREADS; ++s) {
        const int sid  = s * NTHREADS + tid;   // strip id 0..1023
        const int orow = sid >> 7;             // 128 strips per row
        const int c0   = (sid & 127) << 2;     // start column, mult of 4
        const int lr   = orow + 1;             // LDS row of the output row

        float a[3][6], b[3][6];
        #pragma unroll
        for (int i = 0; i < 3; ++i) {
            const float* rp = &sP[(lr - 1 + i) * W];
            const float* rt = &sT[(lr - 1 + i) * W];
            float4 v = *(const float4*)(rp + c0);
            a[i][1] = v.x; a[i][2] = v.y; a[i][3] = v.z; a[i][4] = v.w;
            a[i][0] = (c0 > 0)       ? rp[c0 - 1] : 0.f;
            a[i][5] = (c0 + 4 < W)   ? rp[c0 + 4] : 0.f;
            float4 u = *(const float4*)(rt + c0);
            b[i][1] = u.x; b[i][2] = u.y; b[i][3] = u.z; b[i][4] = u.w;
            b[i][0] = (c0 > 0)       ? rt[c0 - 1] : 0.f;
            b[i][5] = (c0 + 4 < W)   ? rt[c0 + 4] : 0.f;
        }

        #pragma unroll
        for (int j = 0; j < 4; ++j) {
            // Sobel (sign irrelevant under abs)
            float gx = (a[0][j] - a[0][j + 2]) + 2.f * (a[1][j] - a[1][j + 2])
                     + (a[2][j] - a[2][j + 2]);
            float gy = (a[0][j] + 2.f * a[0][j + 1] + a[0][j + 2])
                     - (a[2][j] + 2.f * a[2][j + 1] + a[2][j + 2]);
            float logits = fabsf(gx) + fabsf(gy);

            float gxt = (b[0][j] - b[0][j + 2]) + 2.f * (b[1][j] - b[1][j + 2])
                      + (b[2][j] - b[2][j + 2]);
            float gyt = (b[0][j] + 2.f * b[0][j + 1] + b[0][j + 2])
                      - (b[2][j] + 2.f * b[2][j + 1] + b[2][j + 2]);
            float et = ((fabsf(gxt) + fabsf(gyt)) > 0.f) ? 1.f : 0.f;

            // logits >= 0 : logaddexp(0,x) - x*et = x*(1-et) + log(1+e^-x)
            acc += logits * (1.f - et) + __logf(1.f + __expf(-logits));
        }
    }

    // ---------------- Phase D: deterministic block reduction ------------------
    #pragma unroll
    for (int off = 16; off > 0; off >>= 1)
        acc += __shfl_down(acc, off, 32);           // wave32 shuffle
    const int lane = tid & 31, wid = tid >> 5;
    if (lane == 0) swave[wid] = acc;
    __syncthreads();
    if (tid == 0) {
        float sum = 0.f;
        #pragma unroll
        for (int i = 0; i < NTHREADS / 32; ++i) sum += swave[i];
        partials[blockIdx.x] = sum;
    }
}

__global__ __launch_bounds__(256)
void final_reduce_kernel(const float* __restrict__ partials,
                         float* __restrict__ out) {
    __shared__ float sm[256];
    float s = 0.f;
    for (int i = threadIdx.x; i < NBLOCKS; i += 256) s += partials[i];
    sm[threadIdx.x] = s;
    __syncthreads();
    #pragma unroll
    for (int stride = 128; stride > 0; stride >>= 1) {
        if (threadIdx.x < stride) sm[threadIdx.x] += sm[threadIdx.x + stride];
        __syncthreads();
    }
    if (threadIdx.x == 0)
        out[0] = sm[0] * (1.0f / ((float)BATCH * H * W));
}

extern "C" void kernel_launch(void* const* d_in, const int* in_sizes, int n_in,
                              void* d_out, int out_size, void* d_ws, size_t ws_size,
                              hipStream_t stream) {
    const float* p = (const float*)d_in[0];
    const float* t = (const float*)d_in[1];
    float* partials = (float*)d_ws;        // 4096 floats, rewritten every call

    sobel_bce_tile_kernel<<<NBLOCKS, NTHREADS, 0, stream>>>(p, t, partials);
    final_reduce_kernel<<<1, 256, 0, stream>>>(partials, (float*)d_out);
}